// DApp_49503793053800
// MI455X (gfx1250) — compile-verified
//
#include <hip/hip_runtime.h>

typedef __attribute__((ext_vector_type(2))) float v2f;
typedef __attribute__((ext_vector_type(8))) float v8f;

#define HID      64
#define NCLASS   100
#define NGRAPH   512
#define KOUT     192   // 3 * HID
#define ID_OFF   1500
#define BN_EPS   1e-5f

// ---------------- embedding gather: h[n] = emb[node_ids[n] + 1500] ----------------
__global__ void k_embed(const int* __restrict__ ids, const float* __restrict__ emb,
                        float* __restrict__ h, int n) {
  int t = blockIdx.x * blockDim.x + threadIdx.x;
  if (t >= n * 16) return;
  int node = t >> 4, j = t & 15;
  int id = ids[node] + ID_OFF;
  reinterpret_cast<float4*>(h)[(size_t)node * 16 + j] =
      reinterpret_cast<const float4*>(emb)[(size_t)id * 16 + j];
}

__global__ void k_fill0(float* __restrict__ p, int n) {
  int t = blockIdx.x * blockDim.x + threadIdx.x;
  if (t < n) p[t] = 0.0f;
}

// ---------------- agg init: agg = (1 + eps[l]) * h ----------------
__global__ void k_agg_init(const float* __restrict__ h, float* __restrict__ agg,
                           const float* __restrict__ epsp, int n) {
  int t = blockIdx.x * blockDim.x + threadIdx.x;
  if (t >= n * 16) return;
  float s = 1.0f + *epsp;
  float4 v = reinterpret_cast<const float4*>(h)[t];
  v.x *= s; v.y *= s; v.z *= s; v.w *= s;
  reinterpret_cast<float4*>(agg)[t] = v;
}

// ---------------- edge scatter: agg[dst] += h[src], one wave per edge ----------------
__global__ __launch_bounds__(256)
void k_edge_scatter(const int* __restrict__ src, const int* __restrict__ dst,
                    const float* __restrict__ h, float* __restrict__ agg, int nedges) {
  int e = blockIdx.x * 8 + (threadIdx.x >> 5);
  if (e >= nedges) return;
  int lane = threadIdx.x & 31;
  int s = src[e], d = dst[e];
  float2 v = reinterpret_cast<const float2*>(h + (size_t)s * HID)[lane];
  float* dp = agg + (size_t)d * HID + lane * 2;
  atomicAdd(dp, v.x);
  atomicAdd(dp + 1, v.y);
}

// ---------------- Y = relu(X @ W + b), X:[nrows x 64], W:[64 x 64] ----------------
// One wave computes 16x64 slabs (grid-stride over 16-row tiles).
// W staged in LDS in WMMA-fragment order so every B fragment is one ds_load_b64
// into an even-aligned VGPR pair (lane L -> banks 2L/2L+1, conflict-free).
__global__ __launch_bounds__(256)
void k_mlp_gemm_relu(const float* __restrict__ X, const float* __restrict__ W,
                     const float* __restrict__ B, float* __restrict__ Y, int nrows) {
  __shared__ v2f wfrag[16 * 4 * 32];   // [kb][nt][lane] -> {W[k,col], W[k+1,col]}, 16 KB
  __shared__ float blds[64];
  const int tid = threadIdx.x;
  for (int i = tid; i < 16 * 4 * 32; i += 256) {
    int fl   = i & 31;
    int frag = i >> 5;
    int kbi  = frag >> 2;
    int nt   = frag & 3;
    int k    = kbi * 4 + ((fl >> 4) << 1);   // lanes 16-31 hold k+2 rows (ISA 7.12.2)
    int col  = nt * 16 + (fl & 15);
    v2f w2 = { W[k * HID + col], W[(k + 1) * HID + col] };
    wfrag[i] = w2;
  }
  if (tid < 64) blds[tid] = B[tid];
  __syncthreads();

  const int lane  = tid & 31;
  const int wave  = tid >> 5;
  const int mlane = lane & 15;
  const int khalf = (lane >> 4) << 1;
  const int moff  = (lane >> 4) << 3;      // C/D: rows 8..15 live in lanes 16-31
  const int ntiles = nrows >> 4;           // nrows is a multiple of 16

  for (int tile = blockIdx.x * 8 + wave; tile < ntiles; tile += gridDim.x * 8) {
    const int row0 = tile << 4;
    v8f a0 = {}, a1 = {}, a2 = {}, a3 = {};
    const float* xrow = X + (size_t)(row0 + mlane) * HID;

    for (int kb = 0; kb < 16; ++kb) {
      v2f a = *reinterpret_cast<const v2f*>(xrow + kb * 4 + khalf); // (k,k+1) contiguous
      const v2f* wf = &wfrag[kb * 128 + lane];
      a0 = __builtin_amdgcn_wmma_f32_16x16x4_f32(false, a, false, wf[0],  (short)0, a0, false, false);
      a1 = __builtin_amdgcn_wmma_f32_16x16x4_f32(false, a, false, wf[32], (short)0, a1, false, false);
      a2 = __builtin_amdgcn_wmma_f32_16x16x4_f32(false, a, false, wf[64], (short)0, a2, false, false);
      a3 = __builtin_amdgcn_wmma_f32_16x16x4_f32(false, a, false, wf[96], (short)0, a3, false, false);
    }

    float* yrow = Y + (size_t)(row0 + moff) * HID + mlane;
    #pragma unroll
    for (int r = 0; r < 8; ++r) {
      float v0 = a0[r] + blds[mlane];      v0 = v0 > 0.f ? v0 : 0.f;
      float v1 = a1[r] + blds[16 + mlane]; v1 = v1 > 0.f ? v1 : 0.f;
      float v2 = a2[r] + blds[32 + mlane]; v2 = v2 > 0.f ? v2 : 0.f;
      float v3 = a3[r] + blds[48 + mlane]; v3 = v3 > 0.f ? v3 : 0.f;
      float* yp = yrow + (size_t)r * HID;
      yp[0] = v0; yp[16] = v1; yp[32] = v2; yp[48] = v3;
    }
  }
}

// ---------------- BN stats: sums[0..63]=sum, sums[64..127]=sumsq ----------------
__global__ __launch_bounds__(256)
void k_bn_stats(const float* __restrict__ X, float* __restrict__ sums, int nrows) {
  const int c = threadIdx.x & 63;
  const int g = threadIdx.x >> 6;     // 4 node-groups per block
  const int step = gridDim.x * 4;
  float s = 0.f, s2 = 0.f;
  for (int n = blockIdx.x * 4 + g; n < nrows; n += step) {
    float v = X[(size_t)n * HID + c];
    __builtin_prefetch(&X[(size_t)(n + step) * HID + c], 0, 0);  // global_prefetch_b8
    s += v; s2 += v * v;
  }
  __shared__ float ls[256], ls2[256];
  ls[threadIdx.x] = s; ls2[threadIdx.x] = s2;
  __syncthreads();
  if (g == 0) {
    s  = ls[c]  + ls[64 + c]  + ls[128 + c]  + ls[192 + c];
    s2 = ls2[c] + ls2[64 + c] + ls2[128 + c] + ls2[192 + c];
    atomicAdd(&sums[c], s);
    atomicAdd(&sums[64 + c], s2);
  }
}

__global__ void k_bn_finalize(const float* __restrict__ sums,
                              const float* __restrict__ gamma,
                              const float* __restrict__ beta,
                              float* __restrict__ scsh, int nrows) {
  int c = threadIdx.x;
  if (c >= 64) return;
  float inv_n = 1.0f / (float)nrows;
  float mean = sums[c] * inv_n;
  float var  = sums[64 + c] * inv_n - mean * mean;   // biased var
  float sc   = gamma[c] * rsqrtf(var + BN_EPS);
  scsh[c]      = sc;
  scsh[64 + c] = beta[c] - mean * sc;
}

// ---------------- normalize in place + per-graph pooling ----------------
__global__ __launch_bounds__(256)
void k_bn_apply_pool(float* __restrict__ X, const float* __restrict__ scsh,
                     const int* __restrict__ gids, float* __restrict__ pooled,
                     int n) {
  int t = blockIdx.x * blockDim.x + threadIdx.x;
  if (t >= n * 32) return;
  int node = t >> 5, j = t & 31, c = j * 2;
  float2 v = reinterpret_cast<const float2*>(X + (size_t)node * HID)[j];
  v.x = v.x * scsh[c]     + scsh[64 + c];
  v.y = v.y * scsh[c + 1] + scsh[64 + c + 1];
  reinterpret_cast<float2*>(X + (size_t)node * HID)[j] = v;
  int g = gids[node];
  float* pp = pooled + (size_t)g * KOUT + c;   // pooled already offset by l*64
  atomicAdd(pp, v.x);
  atomicAdd(pp + 1, v.y);
}

// ---------------- out = pooled[512x192] @ W_out[192x100] + b_out ----------------
__global__ __launch_bounds__(32)
void k_out_gemm(const float* __restrict__ P, const float* __restrict__ Wo,
                const float* __restrict__ bo, float* __restrict__ Y) {
  const int lane  = threadIdx.x;
  const int mlane = lane & 15;
  const int khalf = (lane >> 4) << 1;
  const int moff  = (lane >> 4) << 3;
  const int row0  = blockIdx.x * 16;
  const int col   = blockIdx.y * 16 + mlane;
  const bool cok  = (col < NCLASS);            // select, not branch: EXEC stays full

  v8f acc = {};
  const float* prow = P + (size_t)(row0 + mlane) * KOUT;
  for (int kb = 0; kb < KOUT; kb += 4) {
    v2f a = *reinterpret_cast<const v2f*>(prow + kb + khalf);
    v2f b;
    b.x = cok ? Wo[(size_t)(kb + khalf)     * NCLASS + col] : 0.f;
    b.y = cok ? Wo[(size_t)(kb + khalf + 1) * NCLASS + col] : 0.f;
    acc = __builtin_amdgcn_wmma_f32_16x16x4_f32(false, a, false, b, (short)0, acc, false, false);
  }
  if (cok) {
    float bb = bo[col];
    #pragma unroll
    for (int r = 0; r < 8; ++r)
      Y[(size_t)(row0 + moff + r) * NCLASS + col] = acc[r] + bb;
  }
}

// =======================================================================
extern "C" void kernel_launch(void* const* d_in, const int* in_sizes, int n_in,
                              void* d_out, int out_size, void* d_ws, size_t ws_size,
                              hipStream_t stream) {
  const int*   node_ids = (const int*)  d_in[0];
  const int*   edge_src = (const int*)  d_in[1];
  const int*   edge_dst = (const int*)  d_in[2];
  const int*   graph_ids= (const int*)  d_in[3];
  const float* emb      = (const float*)d_in[4];
  const float* Ws       = (const float*)d_in[5];   // [3][3][64][64]
  const float* bs       = (const float*)d_in[6];   // [3][3][64]
  const float* bn_gamma = (const float*)d_in[7];   // [3][64]
  const float* bn_beta  = (const float*)d_in[8];   // [3][64]
  const float* eps      = (const float*)d_in[9];   // [3]
  const float* W_out    = (const float*)d_in[10];  // [192][100]
  const float* b_out    = (const float*)d_in[11];  // [100]
  float* out = (float*)d_out;

  const int N = in_sizes[0];
  const int E = in_sizes[1];
  const size_t NB = (size_t)N * HID;

  float* w      = (float*)d_ws;
  float* buf[3] = { w, w + NB, w + 2 * NB };
  float* pooled = w + 3 * NB;                    // [512][192]
  float* sums   = pooled + (size_t)NGRAPH * KOUT;// [128]
  float* scsh   = sums + 128;                    // [128]

  const int T = 256;
  // h = emb gather ; zero pooled
  k_embed<<<(N * 16 + T - 1) / T, T, 0, stream>>>(node_ids, emb, buf[0], N);
  k_fill0<<<(NGRAPH * KOUT + T - 1) / T, T, 0, stream>>>(pooled, NGRAPH * KOUT);

  const int gemm_blocks = 256;   // grid-stride: ~3 row-tiles per wave, W staged once/block
  int hi = 0;
  for (int l = 0; l < 3; ++l) {
    int ai = (hi + 1) % 3, bi = (hi + 2) % 3;
    // agg = (1+eps)*h + scatter(h[src] -> dst)
    k_agg_init<<<(N * 16 + T - 1) / T, T, 0, stream>>>(buf[hi], buf[ai], eps + l, N);
    k_edge_scatter<<<(E + 7) / 8, T, 0, stream>>>(edge_src, edge_dst, buf[hi], buf[ai], E);
    // 3 x (Linear + ReLU)
    const float* W0 = Ws + ((size_t)l * 3 + 0) * HID * HID;
    const float* W1 = Ws + ((size_t)l * 3 + 1) * HID * HID;
    const float* W2 = Ws + ((size_t)l * 3 + 2) * HID * HID;
    const float* b0 = bs + ((size_t)l * 3 + 0) * HID;
    const float* b1 = bs + ((size_t)l * 3 + 1) * HID;
    const float* b2 = bs + ((size_t)l * 3 + 2) * HID;
    k_mlp_gemm_relu<<<gemm_blocks, T, 0, stream>>>(buf[ai], W0, b0, buf[bi], N);
    k_mlp_gemm_relu<<<gemm_blocks, T, 0, stream>>>(buf[bi], W1, b1, buf[ai], N);
    k_mlp_gemm_relu<<<gemm_blocks, T, 0, stream>>>(buf[ai], W2, b2, buf[bi], N);
    // BatchNorm (training stats) + per-graph pooling
    k_fill0<<<1, 128, 0, stream>>>(sums, 128);
    k_bn_stats<<<512, T, 0, stream>>>(buf[bi], sums, N);
    k_bn_finalize<<<1, 64, 0, stream>>>(sums, bn_gamma + l * HID, bn_beta + l * HID, scsh, N);
    k_bn_apply_pool<<<(N * 32 + T - 1) / T, T, 0, stream>>>(buf[bi], scsh, graph_ids,
                                                            pooled + l * HID, N);
    hi = bi;
  }
  // classifier
  k_out_gemm<<<dim3(NGRAPH / 16, 7), 32, 0, stream>>>(pooled, W_out, b_out, out);
}